// MIA_SS2D_7395933683937
// MI455X (gfx1250) — compile-verified
//
#include <hip/hip_runtime.h>
#include <math.h>

// ---------------- CDNA5 / gfx1250, wave32 ----------------
typedef __attribute__((ext_vector_type(2))) float v2f;
typedef __attribute__((ext_vector_type(8))) float v8f;

constexpr int Bn = 2, Cn = 128, CBn = 32, DIn = 256, NSn = 16, DRn = 8, KDn = 4;
constexpr int Hn = 64, Wn = 64, HWn = Hn * Wn;   // L = 4096

__device__ __forceinline__ float wave_sum(float v) {
#pragma unroll
  for (int o = 16; o > 0; o >>= 1) v += __shfl_xor(v, o, 32);
  return v;
}
__device__ __forceinline__ float wave_max(float v) {
#pragma unroll
  for (int o = 16; o > 0; o >>= 1) v = fmaxf(v, __shfl_xor(v, o, 32));
  return v;
}

// ---------------------------------------------------------------------------
// Generic fp32 WMMA GEMM:  Out[b][m][n] = epi( sum_k A[b][m][k]*(ascale_k) * W[n][k] + bias[n] ) * scalar_b
// A addressed as A + b*a_b + m*a_m + k*a_k (signed strides). W row-major (NxK).
// V_WMMA_F32_16X16X4_F32; EXEC stays all-ones: OOB handled by clamped addresses
// + 0/1 value masks (v_cndmask/v_mul), never by exec-divergent load guards.
// EPI: 0 none, 1 sigmoid, 2 softplus.  AK1: A contiguous in k -> b64 loads.
// ---------------------------------------------------------------------------
template <int EPI, bool AK1, bool HAS_SCALE>
__global__ __launch_bounds__(256) void gemm_wmma_f32(
    const float* __restrict__ A, long a_b, long a_m, long a_k,
    const float* __restrict__ Wt, int w_row,
    const float* __restrict__ bias,
    const float* __restrict__ ascale, long ascale_b,
    const float* __restrict__ bscalar, int bscalar_s,
    float* __restrict__ Out, long o_b, long o_m, long o_n,
    int M, int N, int Kdim) {
  const int tiles_n = (N + 15) >> 4;
  const int tiles_m = (M + 15) >> 4;
  const int tile = blockIdx.x * (blockDim.x >> 5) + (threadIdx.x >> 5);
  if (tile >= tiles_m * tiles_n) return;           // wave-uniform exit
  const int lane = threadIdx.x & 31;
  const int half = lane >> 4, lm = lane & 15;      // ISA 16x4 A layout: lanes 16-31 hold K+2,K+3
  const int tm = tile / tiles_n, tn = tile % tiles_n;
  const int row = tm * 16 + lm, col = tn * 16 + lm;
  const int rowc = (row < M) ? row : (M - 1);      // clamped (valid) addresses
  const int colc = (col < N) ? col : (N - 1);
  const float rmask = (row < M) ? 1.f : 0.f;       // value masks, no EXEC change
  const float cmask = (col < N) ? 1.f : 0.f;

  const float* __restrict__ Ab = A + (long)blockIdx.z * a_b;
  // strength-reduced walking pointers (hoist row/col offsets out of K loop)
  const float* __restrict__ pb = Wt + (long)colc * w_row + (half << 1);
  const float* __restrict__ pa;
  const float* __restrict__ ps = nullptr;
  if (AK1) {
    pa = Ab + (long)rowc * a_m + (half << 1);
    if (HAS_SCALE) ps = ascale + (long)blockIdx.z * ascale_b + (half << 1);
  } else {
    pa = Ab + (long)rowc * a_m + (long)(half << 1) * a_k;
  }

  v8f acc = {0.f, 0.f, 0.f, 0.f, 0.f, 0.f, 0.f, 0.f};
  for (int k0 = 0; k0 < Kdim; k0 += 4) {
    v2f bv = *(const v2f*)pb;                      // rows contiguous in k (8B aligned)
    bv = bv * cmask;
    pb += 4;
    v2f av;
    if (AK1) {
      av = *(const v2f*)pa;
      pa += 4;
      if (HAS_SCALE) {
        v2f sv = *(const v2f*)ps;
        av = av * sv;
        ps += 4;
      }
    } else {
      av.x = pa[0];
      av.y = pa[a_k];
      pa += (a_k << 2);
    }
    av = av * rmask;
    acc = __builtin_amdgcn_wmma_f32_16x16x4_f32(false, av, false, bv, (short)0,
                                                acc, false, false);
  }

  const float smul = bscalar ? bscalar[blockIdx.z * bscalar_s] : 1.f;
  float* Ob = Out + (long)blockIdx.z * o_b;
#pragma unroll
  for (int r = 0; r < 8; ++r) {                    // C/D layout: VGPR r -> rows r, r+8
    const int m = tm * 16 + r + (half << 3);
    const int n = tn * 16 + lm;
    if (m < M && n < N) {
      float v = acc[r];
      if (bias) v += bias[n];
      if (EPI == 1) v = 1.f / (1.f + __expf(-v));
      else if (EPI == 2) v = (v > 20.f) ? v : log1pf(__expf(v));
      Ob[(long)m * o_m + (long)n * o_n] = v * smul;
    }
  }
}

// ---- depthwise 3x3 conv (zero pad = dil), optional bias + SiLU ----
__global__ __launch_bounds__(256) void dwconv3_kernel(
    const float* __restrict__ in, const float* __restrict__ w9,
    const float* __restrict__ bias, float* __restrict__ out, int Cch, int dil,
    int act) {
  long idx = (long)blockIdx.x * 256 + threadIdx.x;
  long total = (long)Bn * Cch * HWn;
  if (idx >= total) return;
  int p = (int)(idx % HWn);
  long bc = idx / HWn;
  int c = (int)(bc % Cch);
  int h = p / Wn, w = p % Wn;
  const float* wp = w9 + c * 9;
  const float* ip = in + bc * HWn;
  float s = 0.f;
#pragma unroll
  for (int dy = -1; dy <= 1; dy++) {
    int hh = h + dy * dil;
    if (hh < 0 || hh >= Hn) continue;
#pragma unroll
    for (int dx = -1; dx <= 1; dx++) {
      int ww = w + dx * dil;
      if (ww < 0 || ww >= Wn) continue;
      s += wp[(dy + 1) * 3 + (dx + 1)] * ip[hh * Wn + ww];
    }
  }
  if (bias) s += bias[c];
  if (act) s = s / (1.f + __expf(-s));
  out[idx] = s;
}

// ---- per-(b,c) spatial mean of x (B,C,H,W) ----
__global__ __launch_bounds__(256) void mean_hw_kernel(const float* __restrict__ x,
                                                      float* __restrict__ gap) {
  const float* p = x + (long)blockIdx.x * HWn;
  float s = 0.f;
  for (int j = threadIdx.x; j < HWn; j += 256) s += p[j];
  s = wave_sum(s);
  __shared__ float sm[8];
  if ((threadIdx.x & 31) == 0) sm[threadIdx.x >> 5] = s;
  __syncthreads();
  if (threadIdx.x == 0) {
    float t = 0.f;
    for (int i = 0; i < 8; i++) t += sm[i];
    gap[blockIdx.x] = t / (float)HWn;
  }
}

// ---- ARF gate: softmax(relu(gap@fc1^T+b1)@fc2^T+b2) ----
__global__ __launch_bounds__(32) void arf_gate_kernel(
    const float* __restrict__ gap, const float* __restrict__ f1w,
    const float* __restrict__ f1b, const float* __restrict__ f2w,
    const float* __restrict__ f2b, float* __restrict__ wgt) {
  int b = blockIdx.x, t = threadIdx.x;
  __shared__ float hid[CBn];
  float a = f1b[t];
  for (int c = 0; c < Cn; c++) a += f1w[t * Cn + c] * gap[b * Cn + c];
  hid[t] = fmaxf(a, 0.f);
  __syncthreads();
  if (t == 0) {
    float lg[4], mx = -3.4e38f;
    for (int i = 0; i < 4; i++) {
      float s = f2b[i];
      for (int j = 0; j < CBn; j++) s += f2w[i * CBn + j] * hid[j];
      lg[i] = s;
      mx = fmaxf(mx, s);
    }
    float den = 0.f;
    for (int i = 0; i < 4; i++) { lg[i] = __expf(lg[i] - mx); den += lg[i]; }
    for (int i = 0; i < 4; i++) wgt[b * 4 + i] = lg[i] / den;
  }
}

// ---- BN batch stats over (B,HW) for e (B,HW,32) ----
__global__ __launch_bounds__(256) void bn_stats_kernel(const float* __restrict__ e,
                                                       float* __restrict__ bm,
                                                       float* __restrict__ bv) {
  int c = blockIdx.x;
  const int Nt = Bn * HWn;
  float s = 0.f, ss = 0.f;
  for (int j = threadIdx.x; j < Nt; j += 256) {
    float v = e[(long)j * CBn + c];
    s += v; ss += v * v;
  }
  s = wave_sum(s); ss = wave_sum(ss);
  __shared__ float sm[8], sq[8];
  if ((threadIdx.x & 31) == 0) { sm[threadIdx.x >> 5] = s; sq[threadIdx.x >> 5] = ss; }
  __syncthreads();
  if (threadIdx.x == 0) {
    float t = 0.f, tq = 0.f;
    for (int i = 0; i < 8; i++) { t += sm[i]; tq += sq[i]; }
    float m = t / (float)Nt;
    bm[c] = m;
    bv[c] = tq / (float)Nt - m * m;
  }
}

__global__ __launch_bounds__(256) void bn_relu_kernel(
    const float* __restrict__ e, const float* __restrict__ bm,
    const float* __restrict__ bv, const float* __restrict__ g,
    const float* __restrict__ bb, float* __restrict__ e2) {
  long idx = (long)blockIdx.x * 256 + threadIdx.x;
  if (idx >= (long)Bn * HWn * CBn) return;
  int c = (int)(idx % CBn);
  float v = (e[idx] - bm[c]) * rsqrtf(bv[c] + 1e-5f) * g[c] + bb[c];
  e2[idx] = fmaxf(v, 0.f);
}

// ---- enhanced = arf_out * (1 + ew) ----
__global__ __launch_bounds__(256) void enhance_kernel(const float* __restrict__ arf,
                                                      const float* __restrict__ ew,
                                                      float* __restrict__ enh) {
  long idx = (long)blockIdx.x * 256 + threadIdx.x;
  if (idx >= (long)Bn * Cn * HWn) return;
  int p = (int)(idx % HWn);
  int c = (int)((idx / HWn) % Cn);
  int b = (int)(idx / ((long)Cn * HWn));
  enh[idx] = arf[idx] * (1.f + ew[((long)b * HWn + p) * Cn + c]);
}

// ---- per-(b,c) mean+max with generic strides ----
__global__ __launch_bounds__(256) void ch_stats_kernel(
    const float* __restrict__ in, long bstr, long cstr, long jstr, int Cch,
    float* __restrict__ outs) {
  int b = blockIdx.x / Cch, c = blockIdx.x % Cch;
  const float* p = in + b * bstr + c * cstr;
  float s = 0.f, mx = -3.4e38f;
  for (int j = threadIdx.x; j < HWn; j += 256) {
    float v = p[(long)j * jstr];
    s += v; mx = fmaxf(mx, v);
  }
  s = wave_sum(s); mx = wave_max(mx);
  __shared__ float sm[8], sx[8];
  if ((threadIdx.x & 31) == 0) { sm[threadIdx.x >> 5] = s; sx[threadIdx.x >> 5] = mx; }
  __syncthreads();
  if (threadIdx.x == 0) {
    float t = 0.f, mm = -3.4e38f;
    for (int i = 0; i < 8; i++) { t += sm[i]; mm = fmaxf(mm, sx[i]); }
    outs[blockIdx.x] = t / (float)HWn + mm;
  }
}

// ---- channel recalibration MLP: sigmoid(relu(s@f1^T)@f2^T) ----
__global__ __launch_bounds__(128) void cr_mlp_kernel(const float* __restrict__ s,
                                                     const float* __restrict__ f1,
                                                     const float* __restrict__ f2,
                                                     float* __restrict__ cw) {
  int b = blockIdx.x, t = threadIdx.x;
  __shared__ float hid[CBn];
  if (t < CBn) {
    float a = 0.f;
    for (int c = 0; c < Cn; c++) a += f1[t * Cn + c] * s[b * Cn + c];
    hid[t] = fmaxf(a, 0.f);
  }
  __syncthreads();
  float a = 0.f;
  for (int j = 0; j < CBn; j++) a += f2[t * CBn + j] * hid[j];
  cw[b * Cn + t] = 1.f / (1.f + __expf(-a));
}

// ---- ss_in = LN_C(arf_out + enhanced*cw), one wave per pixel, 4 ch/lane ----
__global__ __launch_bounds__(256) void ln_ssin_kernel(
    const float* __restrict__ arf, const float* __restrict__ enh,
    const float* __restrict__ cw, const float* __restrict__ g,
    const float* __restrict__ bb, float* __restrict__ out) {
  int wave = threadIdx.x >> 5, lane = threadIdx.x & 31;
  int pix = blockIdx.x * 8 + wave;
  int b = pix / HWn, p = pix % HWn;
  float v[4];
  float s = 0.f, sq = 0.f;
#pragma unroll
  for (int j = 0; j < 4; j++) {
    int c = lane + j * 32;
    long idx = ((long)b * Cn + c) * HWn + p;
    float t = arf[idx] + enh[idx] * cw[b * Cn + c];
    v[j] = t; s += t; sq += t * t;
  }
  s = wave_sum(s); sq = wave_sum(sq);
  float mean = s / (float)Cn;
  float var = sq / (float)Cn - mean * mean;
  float rs = rsqrtf(var + 1e-5f);
#pragma unroll
  for (int j = 0; j < 4; j++) {
    int c = lane + j * 32;
    out[((long)b * HWn + p) * Cn + c] = (v[j] - mean) * rs * g[c] + bb[c];
  }
}

// ---- xt[b,d,l'] = xconv[b,d,(l'%H)*W + l'/H]  (the "wh" scan order) ----
__global__ __launch_bounds__(256) void transpose_hw_kernel(
    const float* __restrict__ in, float* __restrict__ out) {
  long idx = (long)blockIdx.x * 256 + threadIdx.x;
  if (idx >= (long)Bn * DIn * HWn) return;
  int lp = (int)(idx % HWn);
  long bd = idx / HWn;
  out[idx] = in[bd * HWn + (lp % Hn) * Wn + (lp / Hn)];
}

// ---- selective scan: 16 lanes = 16 states per (b,k,d) chain; writes y at the
//      already-inverse-transformed spatial position so directions just sum ----
__global__ __launch_bounds__(256) void scan_kernel(
    const float* __restrict__ xc, const float* __restrict__ xt,
    const float* __restrict__ xdbl, const float* __restrict__ delta,
    const float* __restrict__ Alog, const float* __restrict__ Ds,
    float* __restrict__ ybuf) {
  int gid = blockIdx.x * 256 + threadIdx.x;
  int n = gid & 15;
  int chain = gid >> 4;
  if (chain >= Bn * KDn * DIn) return;
  int d = chain % DIn;
  int k = (chain / DIn) % KDn;
  int b = chain / (DIn * KDn);
  float a = -__expf(Alog[(k * DIn + d) * NSn + n]);
  float dsv = Ds[k * DIn + d];
  const float* usrc = ((k & 1) ? xt : xc) + ((long)b * DIn + d) * HWn;
  bool rev = (k >= 2);
  const float* dp = delta + (((long)(b * KDn + k)) * DIn + d) * (long)HWn;
  const float* bc = xdbl + ((long)(b * KDn + k)) * (long)HWn * 40;
  float* yo = ybuf + (((long)(b * KDn + k)) * DIn + d) * (long)HWn;
  float h = 0.f;
  for (int l = 0; l < HWn; l++) {
    int li = rev ? (HWn - 1 - l) : l;
    float u = usrc[li];
    float dl = dp[l];
    const float* bcl = bc + (long)l * 40;
    float Bv = bcl[8 + n], Cv = bcl[24 + n];
    h = h * __expf(dl * a) + (dl * u) * Bv;
    float yp = h * Cv;
    yp += __shfl_xor(yp, 8, 16);
    yp += __shfl_xor(yp, 4, 16);
    yp += __shfl_xor(yp, 2, 16);
    yp += __shfl_xor(yp, 1, 16);
    if (n == 0) {
      int pos;
      if (k == 0) pos = l;
      else if (k == 2) pos = HWn - 1 - l;
      else { int lp = (k == 1) ? l : (HWn - 1 - l); pos = (lp % Hn) * Wn + (lp / Hn); }
      yo[pos] = yp + u * dsv;
    }
  }
}

// ---- y = sum_k ybuf; LN_DI; * silu(z)  -> g (B,HW,DI) ----
__global__ __launch_bounds__(256) void combine_ln_gate_kernel(
    const float* __restrict__ ybuf, const float* __restrict__ z,
    const float* __restrict__ g, const float* __restrict__ bb,
    float* __restrict__ out) {
  int wave = threadIdx.x >> 5, lane = threadIdx.x & 31;
  int pix = blockIdx.x * 8 + wave;
  int b = pix / HWn, p = pix % HWn;
  const long ks = (long)DIn * HWn;
  float v[8];
  float s = 0.f, sq = 0.f;
#pragma unroll
  for (int j = 0; j < 8; j++) {
    int d = lane + j * 32;
    long base = (((long)b * KDn) * DIn + d) * (long)HWn + p;
    float t = ybuf[base] + ybuf[base + ks] + ybuf[base + 2 * ks] + ybuf[base + 3 * ks];
    v[j] = t; s += t; sq += t * t;
  }
  s = wave_sum(s); sq = wave_sum(sq);
  float mean = s / (float)DIn;
  float var = sq / (float)DIn - mean * mean;
  float rs = rsqrtf(var + 1e-5f);
#pragma unroll
  for (int j = 0; j < 8; j++) {
    int d = lane + j * 32;
    float zz = z[((long)b * HWn + p) * DIn + d];
    out[((long)b * HWn + p) * DIn + d] =
        ((v[j] - mean) * rs * g[d] + bb[d]) * (zz / (1.f + __expf(-zz)));
  }
}

// ---------------------------------------------------------------------------
static void launch_gemm(hipStream_t st, const float* A, long a_b, long a_m,
                        long a_k, const float* Wt, int w_row, const float* bias,
                        const float* ascale, long ascale_b, const float* bscalar,
                        int bscalar_s, float* Out, long o_b, long o_m, long o_n,
                        int M, int N, int Kdim, int epi) {
  int tiles = ((M + 15) / 16) * ((N + 15) / 16);
  dim3 grid((tiles + 7) / 8, 1, Bn);
  dim3 blk(256);
  const bool ak1 = (a_k == 1);
  if (ascale) {  // only used by final projection (AK1, epi 0)
    gemm_wmma_f32<0, true, true><<<grid, blk, 0, st>>>(
        A, a_b, a_m, a_k, Wt, w_row, bias, ascale, ascale_b, bscalar, bscalar_s,
        Out, o_b, o_m, o_n, M, N, Kdim);
  } else if (epi == 1) {
    gemm_wmma_f32<1, true, false><<<grid, blk, 0, st>>>(
        A, a_b, a_m, a_k, Wt, w_row, bias, nullptr, 0, bscalar, bscalar_s, Out,
        o_b, o_m, o_n, M, N, Kdim);
  } else if (epi == 2) {
    gemm_wmma_f32<2, true, false><<<grid, blk, 0, st>>>(
        A, a_b, a_m, a_k, Wt, w_row, bias, nullptr, 0, bscalar, bscalar_s, Out,
        o_b, o_m, o_n, M, N, Kdim);
  } else if (ak1) {
    gemm_wmma_f32<0, true, false><<<grid, blk, 0, st>>>(
        A, a_b, a_m, a_k, Wt, w_row, bias, nullptr, 0, bscalar, bscalar_s, Out,
        o_b, o_m, o_n, M, N, Kdim);
  } else {
    gemm_wmma_f32<0, false, false><<<grid, blk, 0, st>>>(
        A, a_b, a_m, a_k, Wt, w_row, bias, nullptr, 0, bscalar, bscalar_s, Out,
        o_b, o_m, o_n, M, N, Kdim);
  }
}

// workspace layout (floats), with lifetime-based reuse
constexpr long F_DW    = 0;                                  // dwout -> edge -> ew -> ss_in
constexpr long F_CAT   = F_DW + (long)Bn * Cn * HWn;         // cat -> {e, e2} -> enhanced
constexpr long F_E     = F_CAT;
constexpr long F_E2    = F_CAT + (long)Bn * HWn * CBn;
constexpr long F_ARF   = F_CAT + (long)Bn * HWn * Cn;        // arf_out
constexpr long F_XC    = F_ARF + (long)Bn * Cn * HWn;        // xc (cm) -> g
constexpr long F_Z     = F_XC + (long)Bn * DIn * HWn;        // z (pm) -> ss_out
constexpr long F_XCONV = F_Z + (long)Bn * HWn * DIn;
constexpr long F_XT    = F_XCONV + (long)Bn * DIn * HWn;
constexpr long F_XDBL  = F_XT + (long)Bn * DIn * HWn;        // (B,K,L,40)
constexpr long F_DELTA = F_XDBL + (long)Bn * KDn * HWn * 40; // (B,K,DI,L)
constexpr long F_YBUF  = F_DELTA + (long)Bn * KDn * DIn * HWn;
constexpr long F_SM    = F_YBUF + (long)Bn * KDn * DIn * HWn;
constexpr long F_GAP = F_SM, F_WGT = F_GAP + 256, F_BNM = F_WGT + 8,
               F_BNV = F_BNM + 32, F_SPRE = F_BNV + 32, F_CWPRE = F_SPRE + 256,
               F_SPOST = F_CWPRE + 256, F_CWPOST = F_SPOST + 256;

extern "C" void kernel_launch(void* const* d_in, const int* in_sizes, int n_in,
                              void* d_out, int out_size, void* d_ws,
                              size_t ws_size, hipStream_t stream) {
  (void)in_sizes; (void)n_in; (void)out_size; (void)ws_size;
  const float* x          = (const float*)d_in[0];
  const float* arf_dw     = (const float*)d_in[1];
  const float* arf_pw     = (const float*)d_in[2];
  const float* arf_fc1_w  = (const float*)d_in[3];
  const float* arf_fc1_b  = (const float*)d_in[4];
  const float* arf_fc2_w  = (const float*)d_in[5];
  const float* arf_fc2_b  = (const float*)d_in[6];
  const float* arf_fuse_w = (const float*)d_in[7];
  const float* arf_fuse_b = (const float*)d_in[8];
  const float* lee_edge_w = (const float*)d_in[9];
  const float* lee_c1_w   = (const float*)d_in[10];
  const float* lee_c1_b   = (const float*)d_in[11];
  const float* lee_bn_g   = (const float*)d_in[12];
  const float* lee_bn_b   = (const float*)d_in[13];
  const float* lee_c2_w   = (const float*)d_in[14];
  const float* lee_c2_b   = (const float*)d_in[15];
  const float* pre_fc1_w  = (const float*)d_in[16];
  const float* pre_fc2_w  = (const float*)d_in[17];
  const float* ln_g       = (const float*)d_in[18];
  const float* ln_b       = (const float*)d_in[19];
  const float* in_proj_w  = (const float*)d_in[20];
  const float* conv_w     = (const float*)d_in[21];
  const float* conv_b     = (const float*)d_in[22];
  const float* x_proj_w   = (const float*)d_in[23];
  const float* dt_w       = (const float*)d_in[24];
  const float* dt_b       = (const float*)d_in[25];
  const float* A_log      = (const float*)d_in[26];
  const float* Ds         = (const float*)d_in[27];
  const float* out_ln_g   = (const float*)d_in[28];
  const float* out_ln_b   = (const float*)d_in[29];
  const float* out_proj_w = (const float*)d_in[30];
  const float* post_fc1_w = (const float*)d_in[31];
  const float* post_fc2_w = (const float*)d_in[32];
  const float* proj_w     = (const float*)d_in[33];
  const float* proj_b     = (const float*)d_in[34];

  float* ws = (float*)d_ws;
  float* dw = ws + F_DW;       float* cat = ws + F_CAT;
  float* e = ws + F_E;         float* e2 = ws + F_E2;
  float* arf = ws + F_ARF;     float* ew = ws + F_DW;      // reuse
  float* enh = ws + F_CAT;     float* ssin = ws + F_DW;    // reuse
  float* xc = ws + F_XC;       float* zb = ws + F_Z;
  float* xconv = ws + F_XCONV; float* xt = ws + F_XT;
  float* xdbl = ws + F_XDBL;   float* deltab = ws + F_DELTA;
  float* ybuf = ws + F_YBUF;   float* gbuf = ws + F_XC;    // reuse
  float* ssout = ws + F_Z;                                 // reuse
  float* gap = ws + F_GAP;     float* wgt = ws + F_WGT;
  float* bnm = ws + F_BNM;     float* bnv = ws + F_BNV;
  float* spre = ws + F_SPRE;   float* cwpre = ws + F_CWPRE;
  float* spost = ws + F_SPOST; float* cwpost = ws + F_CWPOST;

  const long NE = (long)Bn * Cn * HWn;  // elements of a (B,C,H,W) tensor
  const int dils[4] = {1, 2, 4, 8};

  // --- ARF ---
  mean_hw_kernel<<<Bn * Cn, 256, 0, stream>>>(x, gap);
  arf_gate_kernel<<<Bn, 32, 0, stream>>>(gap, arf_fc1_w, arf_fc1_b, arf_fc2_w,
                                         arf_fc2_b, wgt);
  for (int i = 0; i < 4; i++) {
    dwconv3_kernel<<<(NE + 255) / 256, 256, 0, stream>>>(
        x, arf_dw + (long)i * Cn * 9, nullptr, dw, Cn, dils[i], 0);
    launch_gemm(stream, dw, (long)Cn * HWn, 1, HWn, arf_pw + (long)i * CBn * Cn,
                Cn, nullptr, nullptr, 0, wgt + i, KDn, cat + i * CBn,
                (long)HWn * Cn, Cn, 1, HWn, CBn, Cn, 0);
  }
  launch_gemm(stream, cat, (long)HWn * Cn, Cn, 1, arf_fuse_w, Cn, arf_fuse_b,
              nullptr, 0, nullptr, 0, arf, (long)Cn * HWn, 1, HWn, HWn, Cn, Cn, 0);

  // --- LEE ---
  dwconv3_kernel<<<(NE + 255) / 256, 256, 0, stream>>>(arf, lee_edge_w, nullptr,
                                                       dw, Cn, 1, 0);
  launch_gemm(stream, dw, (long)Cn * HWn, 1, HWn, lee_c1_w, Cn, lee_c1_b,
              nullptr, 0, nullptr, 0, e, (long)HWn * CBn, CBn, 1, HWn, CBn, Cn, 0);
  bn_stats_kernel<<<CBn, 256, 0, stream>>>(e, bnm, bnv);
  bn_relu_kernel<<<((long)Bn * HWn * CBn + 255) / 256, 256, 0, stream>>>(
      e, bnm, bnv, lee_bn_g, lee_bn_b, e2);
  launch_gemm(stream, e2, (long)HWn * CBn, CBn, 1, lee_c2_w, CBn, lee_c2_b,
              nullptr, 0, nullptr, 0, ew, (long)HWn * Cn, Cn, 1, HWn, Cn, CBn, 1);
  enhance_kernel<<<(NE + 255) / 256, 256, 0, stream>>>(arf, ew, enh);

  // --- pre channel recalibration + LN ---
  ch_stats_kernel<<<Bn * Cn, 256, 0, stream>>>(enh, (long)Cn * HWn, HWn, 1, Cn, spre);
  cr_mlp_kernel<<<Bn, 128, 0, stream>>>(spre, pre_fc1_w, pre_fc2_w, cwpre);
  ln_ssin_kernel<<<Bn * HWn / 8, 256, 0, stream>>>(arf, enh, cwpre, ln_g, ln_b, ssin);

  // --- SS2D: in_proj (xc channel-major, z pixel-major), dwconv+silu, transpose ---
  launch_gemm(stream, ssin, (long)HWn * Cn, Cn, 1, in_proj_w, Cn, nullptr,
              nullptr, 0, nullptr, 0, xc, (long)DIn * HWn, 1, HWn, HWn, DIn, Cn, 0);
  launch_gemm(stream, ssin, (long)HWn * Cn, Cn, 1, in_proj_w + (long)DIn * Cn, Cn,
              nullptr, nullptr, 0, nullptr, 0, zb, (long)HWn * DIn, DIn, 1, HWn,
              DIn, Cn, 0);
  dwconv3_kernel<<<((long)Bn * DIn * HWn + 255) / 256, 256, 0, stream>>>(
      xc, conv_w, conv_b, xconv, DIn, 1, 1);
  transpose_hw_kernel<<<((long)Bn * DIn * HWn + 255) / 256, 256, 0, stream>>>(xconv, xt);

  // --- x_proj and dt projection per direction (GEMM views via signed strides) ---
  for (int k = 0; k < 4; k++) {
    const float* asrc = ((k & 1) ? xt : xconv) + ((k >= 2) ? (HWn - 1) : 0);
    launch_gemm(stream, asrc, (long)DIn * HWn, (k >= 2) ? -1 : 1, HWn,
                x_proj_w + (long)k * 40 * DIn, DIn, nullptr, nullptr, 0, nullptr,
                0, xdbl + (long)k * HWn * 40, (long)KDn * HWn * 40, 40, 1, HWn,
                40, DIn, 0);
  }
  for (int k = 0; k < 4; k++) {
    launch_gemm(stream, xdbl + (long)k * HWn * 40, (long)KDn * HWn * 40, 40, 1,
                dt_w + (long)k * DIn * DRn, DRn, dt_b + k * DIn, nullptr, 0,
                nullptr, 0, deltab + (long)k * DIn * HWn, (long)KDn * DIn * HWn,
                1, HWn, HWn, DIn, DRn, 2 /*softplus*/);
  }

  // --- selective scan + combine/LN/gate + out_proj ---
  scan_kernel<<<(Bn * KDn * DIn * 16) / 256, 256, 0, stream>>>(
      xconv, xt, xdbl, deltab, A_log, Ds, ybuf);
  combine_ln_gate_kernel<<<Bn * HWn / 8, 256, 0, stream>>>(ybuf, zb, out_ln_g,
                                                           out_ln_b, gbuf);
  launch_gemm(stream, gbuf, (long)HWn * DIn, DIn, 1, out_proj_w, DIn, nullptr,
              nullptr, 0, nullptr, 0, ssout, (long)HWn * Cn, Cn, 1, HWn, Cn, DIn, 0);

  // --- post channel recalibration + final projection ---
  ch_stats_kernel<<<Bn * Cn, 256, 0, stream>>>(ssout, (long)HWn * Cn, 1, Cn, Cn, spost);
  cr_mlp_kernel<<<Bn, 128, 0, stream>>>(spost, post_fc1_w, post_fc2_w, cwpost);
  launch_gemm(stream, ssout, (long)HWn * Cn, Cn, 1, proj_w, Cn, proj_b, cwpost,
              Cn, nullptr, 0, (float*)d_out, (long)Cn * HWn, 1, HWn, HWn, Cn, Cn, 0);
}